// GraphDecoder_51780125721128
// MI455X (gfx1250) — compile-verified
//
#include <hip/hip_runtime.h>

#define N_NODES 50000
#define N_EDGES 800000
#define IN_DIM  128
#define HID     64
#define GEN_EPS 1e-7f
#define BN_EPS  1e-5f
#define ET_GROUP 4   // edge tiles per wave
#define RT_GROUP 4   // node row tiles per wave

typedef __attribute__((ext_vector_type(2))) float v2f;
typedef __attribute__((ext_vector_type(8))) float v8f;

// D = A(16x4 f32) * B(4x16 f32) + C(16x16 f32)
__device__ __forceinline__ v8f wmma4(v2f a, v2f b, v8f c) {
    return __builtin_amdgcn_wmma_f32_16x16x4_f32(
        /*neg_a=*/false, a, /*neg_b=*/false, b,
        /*c_mod=*/(short)0, c, /*reuse_a=*/false, /*reuse_b=*/false);
}

// ---------------- initial embed: relu(z @ lin_w + lin_b)  [256,128]x[128,64] ----------------
__global__ void embed_kernel(const float* __restrict__ z, const float* __restrict__ w,
                             const float* __restrict__ bias, float* __restrict__ out) {
    int wave = (blockIdx.x * blockDim.x + threadIdx.x) >> 5;
    int lane = threadIdx.x & 31;
    int n = lane & 15, h = lane >> 4;
    int rt = wave >> 2, ct = wave & 3;          // 16 row tiles x 4 col tiles
    int ra = rt * 16 + n;
    v8f acc = {0.f,0.f,0.f,0.f,0.f,0.f,0.f,0.f};
    #pragma unroll
    for (int s = 0; s < IN_DIM / 4; ++s) {
        int k = 4 * s + 2 * h;
        v2f a, b;
        a.x = z[ra * IN_DIM + k];       a.y = z[ra * IN_DIM + k + 1];
        b.x = w[k * HID + ct * 16 + n]; b.y = w[(k + 1) * HID + ct * 16 + n];
        acc = wmma4(a, b, acc);
    }
    int f = ct * 16 + n;
    #pragma unroll
    for (int r = 0; r < 8; ++r) {
        int row = rt * 16 + r + 8 * h;
        out[row * HID + f] = fmaxf(acc[r] + bias[f], 0.0f);
    }
}

// ---------------- gather per-graph latent to nodes ----------------
__global__ void gather_kernel(const float4* __restrict__ embed, const int* __restrict__ batch,
                              float4* __restrict__ x) {
    int i = blockIdx.x * blockDim.x + threadIdx.x;       // over N * (HID/4)
    if (i >= N_NODES * (HID / 4)) return;
    int node = i >> 4, j = i & 15;
    x[i] = embed[batch[node] * (HID / 4) + j];
}

// ---------------- edge passes: msg = relu(x[src] + ea@ew + eb) + eps, then softmax-agg op ----
// MODE 0: mmax[dst,f] = atomic max(msg)     (msgs >= eps > 0; init 0; uint-monotone trick)
// MODE 1: denom[dst,f] += exp(msg - mmax)
// MODE 2: agg[dst,f]  += msg * exp(msg - mmax) / (denom + 1e-16)
template <int MODE>
__global__ void edge_pass_kernel(const float* __restrict__ x,
                                 const int* __restrict__ src, const int* __restrict__ dst,
                                 const float* __restrict__ ea,
                                 const float* __restrict__ ew, const float* __restrict__ eb,
                                 float* __restrict__ mmax, float* __restrict__ denom,
                                 float* __restrict__ agg) {
    int wave = (blockIdx.x * blockDim.x + threadIdx.x) >> 5;
    int lane = threadIdx.x & 31;
    int n = lane & 15, h = lane >> 4;

    // Preload all B fragments of ew (16x64) once per wave: bw[c][s], reused across tiles.
    v2f bw[4][4];
    float ebf[4];
    #pragma unroll
    for (int c = 0; c < 4; ++c) {
        #pragma unroll
        for (int s = 0; s < 4; ++s) {
            int k = 4 * s + 2 * h;
            bw[c][s].x = ew[k * HID + c * 16 + n];
            bw[c][s].y = ew[(k + 1) * HID + c * 16 + n];
        }
        ebf[c] = eb[c * 16 + n];
    }

    #pragma unroll
    for (int t = 0; t < ET_GROUP; ++t) {
        int tile = wave * ET_GROUP + t;
        int e0 = tile * 16;
        if (e0 >= N_EDGES) break;              // wave-uniform (EXEC all-1 around WMMA)

        int sidx[8], didx[8];
        #pragma unroll
        for (int r = 0; r < 8; ++r) {
            int e = e0 + r + 8 * h;
            sidx[r] = src[e];
            didx[r] = dst[e];
        }
        // prefetch gathered source rows (256 B each) while the edge-linear WMMAs run
        #pragma unroll
        for (int r = 0; r < 8; ++r) {
            const float* p = &x[(size_t)sidx[r] * HID];
            __builtin_prefetch(p, 0, 3);
            __builtin_prefetch(p + 32, 0, 3);
        }
        // A fragments (edge_attr row), loaded once, reused across all 4 col tiles
        int ra = e0 + n;
        v2f av[4];
        #pragma unroll
        for (int s = 0; s < 4; ++s) {
            int k = 4 * s + 2 * h;
            av[s].x = ea[ra * 16 + k];
            av[s].y = ea[ra * 16 + k + 1];
        }
        #pragma unroll
        for (int c = 0; c < 4; ++c) {
            v8f acc = {0.f,0.f,0.f,0.f,0.f,0.f,0.f,0.f};
            #pragma unroll
            for (int s = 0; s < 4; ++s) acc = wmma4(av[s], bw[c][s], acc);
            int f = c * 16 + n;
            #pragma unroll
            for (int r = 0; r < 8; ++r) {
                float msg = fmaxf(x[(size_t)sidx[r] * HID + f] + acc[r] + ebf[c], 0.0f) + GEN_EPS;
                int oi = didx[r] * HID + f;
                if (MODE == 0) {
                    atomicMax((unsigned int*)&mmax[oi], __float_as_uint(msg));
                } else {
                    float e = __expf(msg - mmax[oi]);
                    if (MODE == 1) {
                        atomicAdd(&denom[oi], e);
                    } else {
                        float alpha = e / (denom[oi] + 1e-16f);
                        atomicAdd(&agg[oi], msg * alpha);
                    }
                }
            }
        }
    }
}

// ---------------- node MLP stage 1: hmid = (agg + x) @ w1 + b1, accumulate BN stats --------
__global__ void mlp1_kernel(const float* __restrict__ x, const float* __restrict__ agg,
                            const float* __restrict__ w1, const float* __restrict__ b1,
                            float* __restrict__ hmid, float* __restrict__ stats) {
    int wave = (blockIdx.x * blockDim.x + threadIdx.x) >> 5;
    int lane = threadIdx.x & 31;
    int n = lane & 15, h = lane >> 4;
    int ct = wave & 7;                          // 8 col tiles
    int g  = wave >> 3;                         // row-tile group

    v2f bw[16];                                 // B fragments for this col tile (K=64)
    #pragma unroll
    for (int s = 0; s < 16; ++s) {
        int k = 4 * s + 2 * h;
        bw[s].x = w1[k * 128 + ct * 16 + n];
        bw[s].y = w1[(k + 1) * 128 + ct * 16 + n];
    }
    int f = ct * 16 + n;
    float bf = b1[f];

    #pragma unroll
    for (int t = 0; t < RT_GROUP; ++t) {
        int rt = g * RT_GROUP + t;
        if (rt >= N_NODES / 16) break;          // wave-uniform
        int ra = rt * 16 + n;
        v8f acc = {0.f,0.f,0.f,0.f,0.f,0.f,0.f,0.f};
        #pragma unroll
        for (int s = 0; s < 16; ++s) {
            int k = 4 * s + 2 * h;
            v2f a;
            a.x = x[ra * HID + k]     + agg[ra * HID + k];
            a.y = x[ra * HID + k + 1] + agg[ra * HID + k + 1];
            acc = wmma4(a, bw[s], acc);
        }
        float s1 = 0.f, s2 = 0.f;
        #pragma unroll
        for (int r = 0; r < 8; ++r) {
            int row = rt * 16 + r + 8 * h;
            float v = acc[r] + bf;
            hmid[row * 128 + f] = v;
            s1 += v; s2 += v * v;
        }
        s1 += __shfl_xor(s1, 16);
        s2 += __shfl_xor(s2, 16);
        if (lane < 16) {
            atomicAdd(&stats[f], s1);
            atomicAdd(&stats[128 + f], s2);
        }
    }
}

// ---------------- node MLP stage 2: y = relu(bn(hmid)) @ w2 + b2, accumulate BN stats ------
__global__ void mlp2_kernel(const float* __restrict__ hmid,
                            const float* __restrict__ sc1, const float* __restrict__ bi1,
                            const float* __restrict__ w2, const float* __restrict__ b2,
                            float* __restrict__ y, float* __restrict__ stats) {
    int wave = (blockIdx.x * blockDim.x + threadIdx.x) >> 5;
    int lane = threadIdx.x & 31;
    int n = lane & 15, h = lane >> 4;
    int ct = wave & 3;                          // 4 col tiles
    int g  = wave >> 2;                         // row-tile group

    v2f bw[32];                                 // B fragments for this col tile (K=128)
    #pragma unroll
    for (int s = 0; s < 32; ++s) {
        int k = 4 * s + 2 * h;
        bw[s].x = w2[k * HID + ct * 16 + n];
        bw[s].y = w2[(k + 1) * HID + ct * 16 + n];
    }
    int f = ct * 16 + n;
    float bf = b2[f];

    #pragma unroll
    for (int t = 0; t < RT_GROUP; ++t) {
        int rt = g * RT_GROUP + t;
        if (rt >= N_NODES / 16) break;          // wave-uniform
        int ra = rt * 16 + n;
        v8f acc = {0.f,0.f,0.f,0.f,0.f,0.f,0.f,0.f};
        #pragma unroll
        for (int s = 0; s < 32; ++s) {
            int k = 4 * s + 2 * h;
            v2f a;
            a.x = fmaxf(hmid[ra * 128 + k]     * sc1[k]     + bi1[k],     0.f);
            a.y = fmaxf(hmid[ra * 128 + k + 1] * sc1[k + 1] + bi1[k + 1], 0.f);
            acc = wmma4(a, bw[s], acc);
        }
        float s1 = 0.f, s2 = 0.f;
        #pragma unroll
        for (int r = 0; r < 8; ++r) {
            int row = rt * 16 + r + 8 * h;
            float v = acc[r] + bf;
            y[row * HID + f] = v;
            s1 += v; s2 += v * v;
        }
        s1 += __shfl_xor(s1, 16);
        s2 += __shfl_xor(s2, 16);
        if (lane < 16) {
            atomicAdd(&stats[f], s1);
            atomicAdd(&stats[64 + f], s2);
        }
    }
}

// ---------------- BN finalize: per-feature scale/bias from (sum, sumsq) ----------------
__global__ void bn_finalize_kernel(const float* __restrict__ stats, int F, float inv_n,
                                   const float* __restrict__ g, const float* __restrict__ b,
                                   float* __restrict__ scale, float* __restrict__ bias) {
    int f = threadIdx.x;
    if (f < F) {
        float mu  = stats[f] * inv_n;
        float var = stats[F + f] * inv_n - mu * mu;
        float sc  = g[f] * rsqrtf(var + BN_EPS);
        scale[f] = sc;
        bias[f]  = b[f] - mu * sc;
    }
}

// ---------------- apply inter-layer BN (+optional ReLU) ----------------
__global__ void apply_bn_kernel(const float* __restrict__ y,
                                const float* __restrict__ scale, const float* __restrict__ bias,
                                float* __restrict__ out, int relu) {
    int i = blockIdx.x * blockDim.x + threadIdx.x;
    if (i >= N_NODES * HID) return;
    int f = i & (HID - 1);
    float v = y[i] * scale[f] + bias[f];
    if (relu) v = fmaxf(v, 0.f);
    out[i] = v;
}

extern "C" void kernel_launch(void* const* d_in, const int* in_sizes, int n_in,
                              void* d_out, int out_size, void* d_ws, size_t ws_size,
                              hipStream_t stream) {
    const float* z          = (const float*)d_in[0];
    const int*   batch      = (const int*)  d_in[1];
    const int*   edge_index = (const int*)  d_in[2];
    const float* edge_attr  = (const float*)d_in[3];
    const float* lin_w      = (const float*)d_in[4];
    const float* lin_b      = (const float*)d_in[5];
    const int* src = edge_index;
    const int* dst = edge_index + N_EDGES;

    const size_t N64  = (size_t)N_NODES * 64;
    const size_t N128 = (size_t)N_NODES * 128;
    float* ws    = (float*)d_ws;
    float* embed = ws;                 // 256*64
    float* x     = embed + 256 * 64;   // [N,64] layer input
    float* mmax  = x + N64;            // [N,64]  } contiguous:
    float* denom = mmax + N64;         // [N,64]  } single memset
    float* agg   = denom + N64;        // [N,64]  }
    float* hmid  = agg + N64;          // [N,128]
    float* y     = hmid + N128;        // [N,64]
    float* stats = y + N64;            // 256
    float* sc1   = stats + 256;        // 128
    float* bi1   = sc1 + 128;          // 128
    float* sc2   = bi1 + 128;          // 64
    float* bi2   = sc2 + 64;           // 64

    embed_kernel<<<16, 128, 0, stream>>>(z, lin_w, lin_b, embed);
    gather_kernel<<<(N_NODES * 16 + 255) / 256, 256, 0, stream>>>(
        (const float4*)embed, batch, (float4*)x);

    const int edge_waves  = N_EDGES / 16 / ET_GROUP;        // 12500
    const int edge_blocks = edge_waves / 4;                 // 4 waves/block
    const int n_rt        = N_NODES / 16;                   // 3125
    const int rt_groups   = (n_rt + RT_GROUP - 1) / RT_GROUP; // 782
    const int mlp1_blocks = (rt_groups * 8 + 3) / 4;
    const int mlp2_blocks = (rt_groups * 4 + 3) / 4;

    for (int L = 0; L < 3; ++L) {
        const float* ew  = (const float*)d_in[6 + 10 * L + 0];
        const float* eb  = (const float*)d_in[6 + 10 * L + 1];
        const float* w1  = (const float*)d_in[6 + 10 * L + 2];
        const float* b1  = (const float*)d_in[6 + 10 * L + 3];
        const float* bng = (const float*)d_in[6 + 10 * L + 4];
        const float* bnb = (const float*)d_in[6 + 10 * L + 5];
        const float* w2  = (const float*)d_in[6 + 10 * L + 6];
        const float* b2  = (const float*)d_in[6 + 10 * L + 7];
        const float* ng  = (const float*)d_in[6 + 10 * L + 8];
        const float* nb  = (const float*)d_in[6 + 10 * L + 9];

        hipMemsetAsync(mmax, 0, 3 * N64 * sizeof(float), stream);
        edge_pass_kernel<0><<<edge_blocks, 128, 0, stream>>>(x, src, dst, edge_attr, ew, eb, mmax, denom, agg);
        edge_pass_kernel<1><<<edge_blocks, 128, 0, stream>>>(x, src, dst, edge_attr, ew, eb, mmax, denom, agg);
        edge_pass_kernel<2><<<edge_blocks, 128, 0, stream>>>(x, src, dst, edge_attr, ew, eb, mmax, denom, agg);

        hipMemsetAsync(stats, 0, 256 * sizeof(float), stream);
        mlp1_kernel<<<mlp1_blocks, 128, 0, stream>>>(x, agg, w1, b1, hmid, stats);
        bn_finalize_kernel<<<1, 128, 0, stream>>>(stats, 128, 1.0f / N_NODES, bng, bnb, sc1, bi1);

        hipMemsetAsync(stats, 0, 256 * sizeof(float), stream);
        mlp2_kernel<<<mlp2_blocks, 128, 0, stream>>>(hmid, sc1, bi1, w2, b2, y, stats);
        bn_finalize_kernel<<<1, 64, 0, stream>>>(stats, 64, 1.0f / N_NODES, ng, nb, sc2, bi2);

        float* xnext = (L == 2) ? (float*)d_out : x;
        apply_bn_kernel<<<(N_NODES * HID + 255) / 256, 256, 0, stream>>>(y, sc2, bi2, xnext, L < 2 ? 1 : 0);
    }
}